// TopK_with_h_970662609132
// MI455X (gfx1250) — compile-verified
//
#include <hip/hip_runtime.h>
#include <hip/hip_bf16.h>

// Problem constants (match reference)
constexpr int B_ = 256;
constexpr int G_ = 2048;
constexpr int F_ = 128;
constexpr int RNN = 512;
constexpr int K_ = 128;

typedef __attribute__((ext_vector_type(2))) float v2f;
typedef __attribute__((ext_vector_type(8))) float v8f;

// ---------------------------------------------------------------------------
// Kernel 1: scorer = tanh(h_t[256,512] @ W_map[512,128] + b_map)
// One wave per 16x16 output tile, fp32 WMMA 16x16x4, K-loop over 512.
// A 16x4 layout: lane l -> M=l&15, VGPR v -> K = v + 2*(l>>4)
// B 4x16 layout: lane l -> N=l&15, VGPR v -> K = v + 2*(l>>4)
// C/D 16x16:     lane l -> N=l&15, VGPR v -> M = v + 8*(l>>4)
// ---------------------------------------------------------------------------
__global__ __launch_bounds__(32) void mapper_wmma_kernel(
    const float* __restrict__ h_t, const float* __restrict__ W_map,
    const float* __restrict__ b_map, float* __restrict__ scorer) {
  const int lane = threadIdx.x;          // 0..31, one full wave (EXEC all ones)
  const int mn   = lane & 15;            // M for A, N for B/C
  const int khalf = (lane >> 4) * 2;     // 0 or 2
  const int c0 = blockIdx.x * 16;        // output col tile (0..127)
  const int r0 = blockIdx.y * 16;        // output row tile (0..255)

  v8f acc = {0.f, 0.f, 0.f, 0.f, 0.f, 0.f, 0.f, 0.f};

#if defined(__gfx1250__) && __has_builtin(__builtin_amdgcn_wmma_f32_16x16x4_f32)
  const float* arow = h_t + (size_t)(r0 + mn) * RNN;
  for (int k0 = 0; k0 < RNN; k0 += 4) {
    const int ka = k0 + khalf;
    v2f a; a.x = arow[ka];  a.y = arow[ka + 1];
    v2f b; b.x = W_map[(size_t)ka * F_ + c0 + mn];
           b.y = W_map[(size_t)(ka + 1) * F_ + c0 + mn];
    acc = __builtin_amdgcn_wmma_f32_16x16x4_f32(
        /*neg_a=*/false, a, /*neg_b=*/false, b,
        /*c_mod=*/(short)0, acc, /*reuse_a=*/false, /*reuse_b=*/false);
  }
#else
  // Fallback (also taken during host-side parse): direct dot products.
  const int rowhi_f = (lane >> 4) * 8;
  for (int v = 0; v < 8; ++v) {
    const int m = v + rowhi_f;
    float s = 0.f;
    for (int k = 0; k < RNN; ++k)
      s += h_t[(size_t)(r0 + m) * RNN + k] * W_map[(size_t)k * F_ + c0 + mn];
    acc[v] = s;
  }
#endif

  const float bias = b_map[c0 + mn];
  const int rowhi = (lane >> 4) * 8;
#pragma unroll
  for (int v = 0; v < 8; ++v) {
    const int m = v + rowhi;
    scorer[(size_t)(r0 + m) * F_ + c0 + mn] = tanhf(acc[v] + bias);
  }
}

// ---------------------------------------------------------------------------
// Kernel 2: s_hat[b,f] = scorer[b,f] / ||scorer[b,:]||   (one block per row)
// ---------------------------------------------------------------------------
__global__ __launch_bounds__(128) void normalize_kernel(
    const float* __restrict__ scorer, float* __restrict__ shat) {
  __shared__ float red[128];
  const int b = blockIdx.x, t = threadIdx.x;
  const float v = scorer[(size_t)b * F_ + t];
  red[t] = v * v;
  __syncthreads();
  for (int s = 64; s > 0; s >>= 1) {
    if (t < s) red[t] += red[t + s];
    __syncthreads();
  }
  shat[(size_t)b * F_ + t] = v * rsqrtf(red[0]);
}

// ---------------------------------------------------------------------------
// Kernel 3: scores[b,g] = node_embs[b,g,:] . s_hat[b,:] + mask[b,g]
// Memory-bound (256 MB read). One wave per graph row; float4 coalesced loads
// (512B contiguous per wave per load) + wave32 shuffle reduction.
// ---------------------------------------------------------------------------
__global__ __launch_bounds__(256) void scores_kernel(
    const float* __restrict__ node_embs, const float* __restrict__ mask,
    const float* __restrict__ shat, float* __restrict__ scores) {
  const int b = blockIdx.y;
  const int lane = threadIdx.x & 31, wave = threadIdx.x >> 5;
  const float4 sv = reinterpret_cast<const float4*>(shat + (size_t)b * F_)[lane];
  const int gbase = blockIdx.x * 256;
  for (int r = wave; r < 256; r += 8) {
    const int g = gbase + r;
    const float4 e =
        reinterpret_cast<const float4*>(node_embs + ((size_t)b * G_ + g) * F_)[lane];
    float p = e.x * sv.x + e.y * sv.y + e.z * sv.z + e.w * sv.w;
#pragma unroll
    for (int d = 16; d >= 1; d >>= 1) p += __shfl_xor(p, d, 32);
    if (lane == 0)
      scores[(size_t)b * G_ + g] = p + mask[(size_t)b * G_ + g];
  }
}

// ---------------------------------------------------------------------------
// Kernel 4: per batch row: bitonic top-k sort (desc score, asc index),
// logsumexp -> policy score, gather top-128 rows, scale by tanh(val),
// transpose [K,F] -> [F,K] via padded LDS tiles for coalesced stores.
// ---------------------------------------------------------------------------
__global__ __launch_bounds__(256) void topk_kernel(
    const float* __restrict__ node_embs, const float* __restrict__ scores,
    float* __restrict__ out, float* __restrict__ policy) {
  __shared__ float ss[G_];
  __shared__ int   si[G_];
  __shared__ float red[256];
  __shared__ float tile[32][132];  // 16B-aligned rows for float4 LDS stores

  const int b = blockIdx.x, tid = threadIdx.x;

  for (int i = tid; i < G_; i += 256) {
    ss[i] = scores[(size_t)b * G_ + i];
    si[i] = i;
  }
  __syncthreads();

  // Bitonic sort, descending by score, ascending index on ties (== top_k order)
  for (int k = 2; k <= G_; k <<= 1) {
    for (int j = k >> 1; j > 0; j >>= 1) {
      for (int i = tid; i < G_; i += 256) {
        const int ixj = i ^ j;
        if (ixj > i) {
          const float s1 = ss[i], s2 = ss[ixj];
          const int i1 = si[i], i2 = si[ixj];
          const bool firstBefore = (s1 > s2) || (s1 == s2 && i1 < i2);
          const bool desc = ((i & k) == 0);
          if (desc != firstBefore) {
            ss[i] = s2; ss[ixj] = s1;
            si[i] = i2; si[ixj] = i1;
          }
        }
      }
      __syncthreads();
    }
  }

  // logsumexp over all G scores; policy[b] = mean(top-K vals) - logsumexp
  const float mx = ss[0];  // sorted: global max
  float local = 0.f;
  for (int i = tid; i < G_; i += 256) local += expf(ss[i] - mx);
  red[tid] = local;
  __syncthreads();
  for (int s = 128; s > 0; s >>= 1) {
    if (tid < s) red[tid] += red[tid + s];
    __syncthreads();
  }
  if (tid == 0) {
    const float lse = mx + logf(red[0]);
    float msum = 0.f;
    for (int k = 0; k < K_; ++k) msum += ss[k];
    policy[b] = msum * (1.0f / K_) - lse;
  }

  // Gather + scale + transpose: 4 chunks of 32 k-columns
  const int lane = tid & 31, wave = tid >> 5;
  for (int c = 0; c < 4; ++c) {
    for (int r = wave; r < 32; r += 8) {
      const int kk = c * 32 + r;
      const int idx = si[kk];
      const float t = tanhf(ss[kk]);
      const float4 e = reinterpret_cast<const float4*>(
          node_embs + ((size_t)b * G_ + idx) * F_)[lane];
      float4 w = make_float4(e.x * t, e.y * t, e.z * t, e.w * t);
      *reinterpret_cast<float4*>(&tile[r][lane * 4]) = w;
    }
    __syncthreads();
    for (int i = tid; i < F_ * 32; i += 256) {
      const int f = i >> 5, kk = i & 31;
      out[((size_t)b * F_ + f) * K_ + c * 32 + kk] = tile[kk][f];
    }
    __syncthreads();
  }
}

// ---------------------------------------------------------------------------
extern "C" void kernel_launch(void* const* d_in, const int* in_sizes, int n_in,
                              void* d_out, int out_size, void* d_ws, size_t ws_size,
                              hipStream_t stream) {
  (void)in_sizes; (void)n_in; (void)out_size; (void)ws_size;
  const float* node_embs = (const float*)d_in[0];  // [B,G,F]
  const float* mask      = (const float*)d_in[1];  // [B,G]
  const float* h_t       = (const float*)d_in[2];  // [B,RNN]
  const float* W_map     = (const float*)d_in[3];  // [RNN,F]
  const float* b_map     = (const float*)d_in[4];  // [F]

  float* out    = (float*)d_out;                       // [B,F,K] flat
  float* policy = out + (size_t)B_ * F_ * K_;          // [B]

  float* scorer = (float*)d_ws;                        // [B,F]
  float* shat   = scorer + (size_t)B_ * F_;            // [B,F]
  float* scores = shat + (size_t)B_ * F_;              // [B,G]

  mapper_wmma_kernel<<<dim3(F_ / 16, B_ / 16), 32, 0, stream>>>(h_t, W_map, b_map, scorer);
  normalize_kernel<<<B_, 128, 0, stream>>>(scorer, shat);
  scores_kernel<<<dim3(G_ / 256, B_), 256, 0, stream>>>(node_embs, mask, shat, scores);
  topk_kernel<<<B_, 256, 0, stream>>>(node_embs, scores, out, policy);
}